// ReEig_78778290143326
// MI455X (gfx1250) — compile-verified
//
#include <hip/hip_runtime.h>
#include <math.h>

// ReEig for B=16384 symmetric 64x64 fp32 matrices:
//   f(X) = 0.5*(X + eps*I) + 0.5*|X - eps*I|,  |M| = M * sign(M)
// sign(M) via Polar-Express quintic + Newton-Schulz cubic iterations:
// pure 64x64x64 fp32 matmuls on V_WMMA_F32_16X16X4_F32, operands in LDS.
//
// LDS layout: row-padded (LD=66 floats) -> affine addresses (one base VGPR
// + immediate ds offsets), 8B row alignment (A-frag pairs -> ds_load_b64),
// and conflict-free stride-66 column reads ((2r+c) mod 64 distinct).
// Inner loop is software-pipelined: kk+1 fragments prefetched during kk WMMAs.

typedef __attribute__((ext_vector_type(2))) float v2f;
typedef __attribute__((ext_vector_type(8))) float v8f;

#define EPSILON 1e-6f
#define N64 64
#define NELEM (N64 * N64)
#define LDS_LD 66
#define LDS_ELEMS (N64 * LDS_LD)

__device__ __forceinline__ int ldx(int r, int c) { return r * LDS_LD + c; }

__device__ __forceinline__ v2f wmma4(v2f a, v2f b, v8f& acc) {
    acc = __builtin_amdgcn_wmma_f32_16x16x4_f32(false, a, false, b,
                                                (short)0, acc, false, false);
    return a;
}

// One 64x64x64 fp32 matmul, 4 waves cooperate; wave w computes rows [16w,16w+16).
//   D = alpha*(A @ B) + c0s*C0 + c1s*C1     (C0/C1 optional, elementwise)
// In-place D==A / D==C1 is safe: each wave reads only its own row band of
// those operands and all its LDS reads precede its writes (per-wave DS order).
__device__ __forceinline__ void mm64(const float* Ab, const float* Bb,
                                     float* Db, float alpha,
                                     const float* C0, float c0s,
                                     const float* C1, float c1s,
                                     int wave, int lane) {
    const int half = lane >> 4;   // 0: lanes 0-15, 1: lanes 16-31
    const int l = lane & 15;
    const int mrow0 = wave * 16;

    v8f acc[4];
#pragma unroll
    for (int n = 0; n < 4; ++n) {
        v8f c = {};
        if (C0) {
#pragma unroll
            for (int r = 0; r < 8; ++r) {
                const int row = mrow0 + r + half * 8;   // C/D VGPR layout
                const int col = n * 16 + l;
                float v = c0s * C0[ldx(row, col)];
                if (C1) v += c1s * C1[ldx(row, col)];
                c[r] = v;
            }
        }
        acc[n] = c;
    }

    const int ar = mrow0 + l;
    const int kh = half * 2;      // A: halves hold K 0,1 / 2,3;  B: rows K 0,1 / 2,3

    // --- software-pipelined k-loop: prefetch kk+1 while issuing kk WMMAs ---
    v2f af, bf0, bf1, bf2, bf3;
    af.x = Ab[ldx(ar, kh)];     af.y = Ab[ldx(ar, kh + 1)];       // ds_load_b64
    bf0.x = Bb[ldx(kh, l)];          bf0.y = Bb[ldx(kh + 1, l)];
    bf1.x = Bb[ldx(kh, 16 + l)];     bf1.y = Bb[ldx(kh + 1, 16 + l)];
    bf2.x = Bb[ldx(kh, 32 + l)];     bf2.y = Bb[ldx(kh + 1, 32 + l)];
    bf3.x = Bb[ldx(kh, 48 + l)];     bf3.y = Bb[ldx(kh + 1, 48 + l)];

#pragma unroll
    for (int kk = 0; kk < 16; ++kk) {
        v2f afn = af, b0n = bf0, b1n = bf1, b2n = bf2, b3n = bf3;
        if (kk < 15) {
            const int kb = (kk + 1) * 4 + kh;
            afn.x = Ab[ldx(ar, kb)];   afn.y = Ab[ldx(ar, kb + 1)];
            b0n.x = Bb[ldx(kb, l)];        b0n.y = Bb[ldx(kb + 1, l)];
            b1n.x = Bb[ldx(kb, 16 + l)];   b1n.y = Bb[ldx(kb + 1, 16 + l)];
            b2n.x = Bb[ldx(kb, 32 + l)];   b2n.y = Bb[ldx(kb + 1, 32 + l)];
            b3n.x = Bb[ldx(kb, 48 + l)];   b3n.y = Bb[ldx(kb + 1, 48 + l)];
        }
        v2f as;
        as.x = alpha * af.x;
        as.y = alpha * af.y;
        wmma4(as, bf0, acc[0]);
        wmma4(as, bf1, acc[1]);
        wmma4(as, bf2, acc[2]);
        wmma4(as, bf3, acc[3]);
        af = afn; bf0 = b0n; bf1 = b1n; bf2 = b2n; bf3 = b3n;
    }

#pragma unroll
    for (int n = 0; n < 4; ++n)
#pragma unroll
        for (int r = 0; r < 8; ++r) {
            const int row = mrow0 + r + half * 8;
            const int col = n * 16 + l;
            Db[ldx(row, col)] = acc[n][r];
        }
}

// Final: Out = 0.5*Xraw + 0.5*eps*I + (0.5*(Xraw - eps*I)) @ S, streamed to HBM.
// Xraw lives (re-staged) in LDS buffer Xb_lds; S is the converged sign iterate.
__device__ __forceinline__ void mm64_out(const float* Xl, const float* S,
                                         float* __restrict__ out,
                                         int wave, int lane) {
    const int half = lane >> 4;
    const int l = lane & 15;
    const int mrow0 = wave * 16;

    v8f acc[4];
#pragma unroll
    for (int n = 0; n < 4; ++n) {
        v8f c = {};
#pragma unroll
        for (int r = 0; r < 8; ++r) {
            const int row = mrow0 + r + half * 8;
            const int col = n * 16 + l;
            float v = 0.5f * Xl[ldx(row, col)];
            if (row == col) v += 0.5f * EPSILON;
            c[r] = v;
        }
        acc[n] = c;
    }

    const int ar = mrow0 + l;
    const int kh = half * 2;

    v2f af, bf0, bf1, bf2, bf3;
    af.x = Xl[ldx(ar, kh)];     af.y = Xl[ldx(ar, kh + 1)];
    bf0.x = S[ldx(kh, l)];          bf0.y = S[ldx(kh + 1, l)];
    bf1.x = S[ldx(kh, 16 + l)];     bf1.y = S[ldx(kh + 1, 16 + l)];
    bf2.x = S[ldx(kh, 32 + l)];     bf2.y = S[ldx(kh + 1, 32 + l)];
    bf3.x = S[ldx(kh, 48 + l)];     bf3.y = S[ldx(kh + 1, 48 + l)];

#pragma unroll
    for (int kk = 0; kk < 16; ++kk) {
        const int kb = kk * 4 + kh;
        v2f afn = af, b0n = bf0, b1n = bf1, b2n = bf2, b3n = bf3;
        if (kk < 15) {
            const int kp = kb + 4;
            afn.x = Xl[ldx(ar, kp)];   afn.y = Xl[ldx(ar, kp + 1)];
            b0n.x = S[ldx(kp, l)];        b0n.y = S[ldx(kp + 1, l)];
            b1n.x = S[ldx(kp, 16 + l)];   b1n.y = S[ldx(kp + 1, 16 + l)];
            b2n.x = S[ldx(kp, 32 + l)];   b2n.y = S[ldx(kp + 1, 32 + l)];
            b3n.x = S[ldx(kp, 48 + l)];   b3n.y = S[ldx(kp + 1, 48 + l)];
        }
        v2f as;   // A-frag = 0.5*(X - eps*I)
        as.x = 0.5f * (af.x - ((ar == kb)     ? EPSILON : 0.0f));
        as.y = 0.5f * (af.y - ((ar == kb + 1) ? EPSILON : 0.0f));
        wmma4(as, bf0, acc[0]);
        wmma4(as, bf1, acc[1]);
        wmma4(as, bf2, acc[2]);
        wmma4(as, bf3, acc[3]);
        af = afn; bf0 = b0n; bf1 = b1n; bf2 = b2n; bf3 = b3n;
    }

#pragma unroll
    for (int n = 0; n < 4; ++n)
#pragma unroll
        for (int r = 0; r < 8; ++r) {
            const int row = mrow0 + r + half * 8;
            const int col = n * 16 + l;
            out[row * N64 + col] = acc[n][r];
        }
}

__global__ void __launch_bounds__(128) reeig64_kernel(const float* __restrict__ X,
                                                      float* __restrict__ Out) {
    __shared__ float sY[LDS_ELEMS];   // sign iterate
    __shared__ float sT[LDS_ELEMS];   // Y^2 (also staging for X)
    __shared__ float sP[LDS_ELEMS];   // Y^3 / next iterate (sP[0]: ssq scratch)

    const int b = blockIdx.x;
    const float* Xg = X + (size_t)b * NELEM;
    float* Ob = Out + (size_t)b * NELEM;
    const int t = threadIdx.x;      // 0..127
    const int lane = t & 31;
    const int wave = t >> 5;

    if (t == 0) sP[0] = 0.0f;
    __syncthreads();

    // ---- Stage 1: load X, shift diagonal by -eps into sT, Frobenius^2 ----
    float ssq = 0.0f;
#pragma unroll
    for (int i = 0; i < 8; ++i) {
        const int v4 = i * 128 + t;          // float4 index (0..1023)
        const float4 q = reinterpret_cast<const float4*>(Xg)[v4];
        const int r = v4 >> 4;               // 16 float4 per row
        const int c = (v4 & 15) * 4;
        float e[4] = {q.x, q.y, q.z, q.w};
        const int d = r - c;
        if (d >= 0 && d < 4) e[d] -= EPSILON;
        ssq += e[0] * e[0] + e[1] * e[1] + e[2] * e[2] + e[3] * e[3];
        const int o = ldx(r, c);
        sT[o + 0] = e[0]; sT[o + 1] = e[1]; sT[o + 2] = e[2]; sT[o + 3] = e[3];
    }
    atomicAdd(&sP[0], ssq);                  // ds_add_f32
    __syncthreads();
    const float s = sqrtf(sP[0]) + 1e-30f;   // Frobenius >= spectral norm
    const float inv = 1.0f / s;
    __syncthreads();                         // all reads of sP[0] done

    // ---- Stage 2: Y = (X - eps*I)/s ----
#pragma unroll
    for (int i = 0; i < 8; ++i) {
        const int v4 = i * 128 + t;
        const int o = ldx(v4 >> 4, (v4 & 15) * 4);
#pragma unroll
        for (int j = 0; j < 4; ++j) sY[o + j] = sT[o + j] * inv;
    }
    __syncthreads();

    // ---- Stage 3: S = sign(Y0) ----
    // 9 aggressive quintic (Polar-Express) iterations: Y <- aY + bY^3 + cY^5
    const float qa = 3.4445f, qb = -4.7750f, qc = 2.0315f;
    float* Y = sY;
    float* P = sP;
#pragma unroll 1
    for (int it = 0; it < 9; ++it) {
        mm64(Y, Y, sT, 1.0f, nullptr, 0.0f, nullptr, 0.0f, wave, lane);  // T = Y^2
        __syncthreads();
        mm64(Y, sT, P, 1.0f, nullptr, 0.0f, nullptr, 0.0f, wave, lane);  // P = Y^3
        __syncthreads();
        mm64(P, sT, P, qc, Y, qa, P, qb, wave, lane);  // P = qa*Y + qb*Y^3 + qc*Y^5
        __syncthreads();
        float* tmp = Y; Y = P; P = tmp;
    }
    // 4 Newton-Schulz cubic polish iterations: Y <- 1.5Y - 0.5Y^3 (quadratic at ±1)
#pragma unroll 1
    for (int it = 0; it < 4; ++it) {
        mm64(Y, Y, sT, 1.0f, nullptr, 0.0f, nullptr, 0.0f, wave, lane);  // T = Y^2
        __syncthreads();
        mm64(Y, sT, Y, -0.5f, Y, 1.5f, nullptr, 0.0f, wave, lane);       // in-place band
        __syncthreads();
    }

    // ---- Stage 4: re-stage raw X into sT, then Out = 0.5(X + eps I) + 0.5(X-eps I)S
#pragma unroll
    for (int i = 0; i < 8; ++i) {
        const int v4 = i * 128 + t;
        const float4 q = reinterpret_cast<const float4*>(Xg)[v4];
        const int o = ldx(v4 >> 4, (v4 & 15) * 4);
        sT[o + 0] = q.x; sT[o + 1] = q.y; sT[o + 2] = q.z; sT[o + 3] = q.w;
    }
    __syncthreads();
    mm64_out(sT, Y, Ob, wave, lane);
}

extern "C" void kernel_launch(void* const* d_in, const int* in_sizes, int n_in,
                              void* d_out, int out_size, void* d_ws, size_t ws_size,
                              hipStream_t stream) {
    const float* X = (const float*)d_in[0];
    float* out = (float*)d_out;
    const int nb = in_sizes[0] / NELEM;      // 16384 matrices
    dim3 grid(nb), block(128);
    hipLaunchKernelGGL(reeig64_kernel, grid, block, 0, stream, X, out);
}